// GNNWithDenseDiffPool_21380347199506
// MI455X (gfx1250) — compile-verified
//
#include <hip/hip_runtime.h>
#include <hip/hip_bf16.h>

typedef __attribute__((ext_vector_type(2))) float v2f;
typedef __attribute__((ext_vector_type(8))) float v8f;

__device__ __forceinline__ void atomAddF(float* p, float v) {
  __hip_atomic_fetch_add(p, v, __ATOMIC_RELAXED, __HIP_MEMORY_SCOPE_AGENT);
}

// ---------------- CSR construction (once per call; graph shared by all layers) ---------
__global__ void k_zero_int(int* p, int n) {
  int i = blockIdx.x * blockDim.x + threadIdx.x;
  if (i < n) p[i] = 0;
}
__global__ void k_indeg(const int* __restrict__ dst, int* __restrict__ cnt, int e) {
  int i = blockIdx.x * blockDim.x + threadIdx.x;
  if (i < e) atomicAdd(&cnt[dst[i]], 1);
}
__global__ void k_dinv(const int* __restrict__ cnt, float* __restrict__ dinv, int n) {
  int i = blockIdx.x * blockDim.x + threadIdx.x;
  if (i < n) dinv[i] = rsqrtf((float)(cnt[i] + 1));  // +1 = self loop
}
// single-block exclusive scan over n counts -> off[0..n]
__global__ __launch_bounds__(1024) void k_scan(const int* __restrict__ cnt,
                                               int* __restrict__ off, int n) {
  __shared__ int temp[1024];
  __shared__ int carry;
  if (threadIdx.x == 0) carry = 0;
  __syncthreads();
  for (int base = 0; base < n; base += 1024) {
    int i = base + threadIdx.x;
    int v = (i < n) ? cnt[i] : 0;
    temp[threadIdx.x] = v;
    __syncthreads();
#pragma unroll
    for (int d = 1; d < 1024; d <<= 1) {
      int t = (threadIdx.x >= d) ? temp[threadIdx.x - d] : 0;
      __syncthreads();
      temp[threadIdx.x] += t;
      __syncthreads();
    }
    int incl = temp[threadIdx.x];
    if (i < n) off[i] = carry + incl - v;  // exclusive
    __syncthreads();
    if (threadIdx.x == 1023) carry += temp[1023];
    __syncthreads();
  }
  if (threadIdx.x == 0) off[n] = carry;
}
__global__ void k_copy_int(const int* __restrict__ a, int* __restrict__ b, int n) {
  int i = blockIdx.x * blockDim.x + threadIdx.x;
  if (i < n) b[i] = a[i];
}
__global__ void k_fill_csr(const int* __restrict__ src, const int* __restrict__ dst,
                           int* __restrict__ cursor, int* __restrict__ csr, int e) {
  int i = blockIdx.x * blockDim.x + threadIdx.x;
  if (i < e) {
    int d = dst[i];
    int pos = atomicAdd(&cursor[d], 1);
    csr[pos] = src[i];
  }
}

// ---------------- pad head weights to 128 ----------------
__global__ void k_pad(const float* __restrict__ Wl1, const float* __restrict__ bl1,
                      const float* __restrict__ Wl2,
                      float* Wl1p, float* bl1p, float* Wl2p) {
  int i = blockIdx.x * blockDim.x + threadIdx.x;
  if (i >= 128 * 128) return;
  int r = i >> 7, c = i & 127;
  Wl1p[i] = (c < 125) ? Wl1[r * 125 + c] : 0.0f;   // Wl1: 128x125 -> 128x128
  Wl2p[i] = (r < 125) ? Wl2[i] : 0.0f;             // Wl2: 125x128 -> 128x128
  if (i < 128) bl1p[i] = (i < 125) ? bl1[i] : 0.0f;
}

// ---------------- WMMA GEMM: Y[n x 128] = X[n x 128] @ W[128 x 128] ----------------
__global__ __launch_bounds__(256) void k_gemm128(const float* __restrict__ X,
                                                 const float* __restrict__ W,
                                                 float* __restrict__ Y, int n) {
  __shared__ float sA[8][16 * 128];
  const int tid  = threadIdx.x;
  const int wave = tid >> 5;
  const int lane = tid & 31;
  const int row0 = blockIdx.x * 128 + wave * 16;

  for (int i = lane; i < 16 * 128; i += 32) {
    int r = i >> 7, c = i & 127;
    int gr = row0 + r;
    sA[wave][i] = (gr < n) ? X[(size_t)gr * 128 + c] : 0.0f;
  }
  __syncthreads();

  const int mn = lane & 15;
  const int kb = (lane >> 4) << 1;
  v8f acc[8];
#pragma unroll
  for (int nc = 0; nc < 8; ++nc) acc[nc] = (v8f){0,0,0,0,0,0,0,0};

  for (int k0 = 0; k0 < 128; k0 += 4) {
    v2f a;
    a.x = sA[wave][mn * 128 + k0 + kb];
    a.y = sA[wave][mn * 128 + k0 + kb + 1];
#pragma unroll
    for (int nc = 0; nc < 8; ++nc) {
      v2f b;
      b.x = W[(size_t)(k0 + kb) * 128 + nc * 16 + mn];
      b.y = W[(size_t)(k0 + kb + 1) * 128 + nc * 16 + mn];
      acc[nc] = __builtin_amdgcn_wmma_f32_16x16x4_f32(
          false, a, false, b, (short)0, acc[nc], false, false);
    }
  }

  const int mbase = (lane >> 4) * 8;
#pragma unroll
  for (int nc = 0; nc < 8; ++nc) {
#pragma unroll
    for (int r = 0; r < 8; ++r) {
      int gr = row0 + mbase + r;
      if (gr < n) Y[(size_t)gr * 128 + nc * 16 + mn] = acc[nc][r];
    }
  }
}

// -------- gather aggregation: wave per node, fused self-loop + bias + relu ----------
// Out[i,:] = act( dinv[i]^2*T[i,:] + sum_{s in csr[off[i]:off[i+1]]} dinv[i]*dinv[s]*T[s,:] + b )
__global__ __launch_bounds__(256) void k_aggregate(const float* __restrict__ T,
                                                   float* __restrict__ Out,
                                                   const int* __restrict__ csr,
                                                   const int* __restrict__ off,
                                                   const float* __restrict__ dinv,
                                                   const float* __restrict__ bias,
                                                   int n, int relu) {
  int node = blockIdx.x * 8 + (threadIdx.x >> 5);
  int lane = threadIdx.x & 31;
  if (node >= n) return;
  float di = dinv[node];
  float4 v = ((const float4*)(T + (size_t)node * 128))[lane];
  float w = di * di;
  float4 acc;
  acc.x = v.x * w; acc.y = v.y * w; acc.z = v.z * w; acc.w = v.w * w;
  int j0 = off[node], j1 = off[node + 1];
  for (int j = j0; j < j1; ++j) {
    int s = csr[j];                       // uniform per wave
    float nrm = di * dinv[s];
    float4 t = ((const float4*)(T + (size_t)s * 128))[lane];  // 512B coalesced gather
    acc.x += nrm * t.x; acc.y += nrm * t.y;
    acc.z += nrm * t.z; acc.w += nrm * t.w;
  }
  int jc = lane * 4;
  acc.x += bias[jc]; acc.y += bias[jc + 1]; acc.z += bias[jc + 2]; acc.w += bias[jc + 3];
  if (relu) {
    acc.x = fmaxf(acc.x, 0.f); acc.y = fmaxf(acc.y, 0.f);
    acc.z = fmaxf(acc.z, 0.f); acc.w = fmaxf(acc.w, 0.f);
  }
  ((float4*)(Out + (size_t)node * 128))[lane] = acc;
}

// ---------------- bias (+optional residual) (+relu), in place ----------------
__global__ void k_bias_act(float* __restrict__ A, const float* __restrict__ b,
                           const float* __restrict__ res, int n, int relu) {
  int idx = blockIdx.x * blockDim.x + threadIdx.x;
  if (idx >= n * 32) return;
  int j = (idx & 31) * 4;
  float4 v = ((float4*)A)[idx];
  v.x += b[j]; v.y += b[j + 1]; v.z += b[j + 2]; v.w += b[j + 3];
  if (res) {
    float4 r = ((const float4*)res)[idx];
    v.x += r.x; v.y += r.y; v.z += r.z; v.w += r.w;
  }
  if (relu) {
    v.x = fmaxf(v.x, 0.f); v.y = fmaxf(v.y, 0.f);
    v.z = fmaxf(v.z, 0.f); v.w = fmaxf(v.w, 0.f);
  }
  ((float4*)A)[idx] = v;
}

// ---------------- head (128->2) + mean-pool, wave per node ----------------
__global__ void k_zero(float* p, int n) {
  int i = blockIdx.x * blockDim.x + threadIdx.x;
  if (i < n) p[i] = 0.0f;
}
__global__ __launch_bounds__(256) void k_head_pool(const float* __restrict__ H2,
                                                   const float* __restrict__ Wl3,
                                                   const float* __restrict__ bl3,
                                                   const int* __restrict__ batch,
                                                   float* __restrict__ sums,
                                                   float* __restrict__ cnt, int n) {
  int node = blockIdx.x * 8 + (threadIdx.x >> 5);
  int lane = threadIdx.x & 31;
  if (node >= n) return;
  float4 v = ((const float4*)(H2 + (size_t)node * 128))[lane];
  int k0 = lane * 4;
  float d0 = v.x * Wl3[(k0+0)*2+0] + v.y * Wl3[(k0+1)*2+0]
           + v.z * Wl3[(k0+2)*2+0] + v.w * Wl3[(k0+3)*2+0];
  float d1 = v.x * Wl3[(k0+0)*2+1] + v.y * Wl3[(k0+1)*2+1]
           + v.z * Wl3[(k0+2)*2+1] + v.w * Wl3[(k0+3)*2+1];
#pragma unroll
  for (int off = 16; off; off >>= 1) {
    d0 += __shfl_down(d0, off, 32);
    d1 += __shfl_down(d1, off, 32);
  }
  if (lane == 0) {
    float o0 = fmaxf(d0 + bl3[0], 0.f);
    float o1 = fmaxf(d1 + bl3[1], 0.f);
    int g = batch[node];
    atomAddF(&sums[g * 2 + 0], o0);
    atomAddF(&sums[g * 2 + 1], o1);
    atomAddF(&cnt[g], 1.0f);
  }
}
__global__ void k_mean(const float* __restrict__ sums, const float* __restrict__ cnt,
                       float* __restrict__ out, int g2) {
  int i = blockIdx.x * blockDim.x + threadIdx.x;
  if (i < g2) out[i] = sums[i] / fmaxf(cnt[i >> 1], 1.0f);
}

extern "C" void kernel_launch(void* const* d_in, const int* in_sizes, int n_in,
                              void* d_out, int out_size, void* d_ws, size_t ws_size,
                              hipStream_t stream) {
  const float* x    = (const float*)d_in[0];
  const int*   ei   = (const int*)d_in[1];
  const int*   batch= (const int*)d_in[2];
  const float* W1   = (const float*)d_in[3];
  const float* b1   = (const float*)d_in[4];
  const float* W2   = (const float*)d_in[5];
  const float* b2   = (const float*)d_in[6];
  const float* W3   = (const float*)d_in[7];
  const float* b3   = (const float*)d_in[8];
  const float* Wl1  = (const float*)d_in[9];
  const float* bl1  = (const float*)d_in[10];
  const float* Wl2  = (const float*)d_in[11];
  const float* bl2  = (const float*)d_in[12];
  const float* Wl3  = (const float*)d_in[13];
  const float* bl3  = (const float*)d_in[14];
  float* out = (float*)d_out;

  const int N = in_sizes[0] / 128;
  const int E = in_sizes[1] / 2;
  const int G = out_size / 2;
  const int* src = ei;
  const int* dst = ei + E;

  // workspace layout
  float* ws = (float*)d_ws;
  size_t nf = (size_t)N * 128;
  float* T    = ws;
  float* A1   = ws + nf;
  float* A2   = ws + 2 * nf;
  float* dinv = ws + 3 * nf;        // N
  float* Wl1p = dinv + N;           // 16384
  float* bl1p = Wl1p + 16384;       // 128
  float* Wl2p = bl1p + 128;         // 16384
  float* sums = Wl2p + 16384;       // 2G
  float* cnt  = sums + 2 * G;       // G
  int* indeg  = (int*)(cnt + G);    // N
  int* off    = indeg + N;          // N+1
  int* cursor = off + N + 1;        // N
  int* csr    = cursor + N;         // E
  (void)ws_size; (void)n_in;

  const int TB = 256;
  dim3 blkN((N + TB - 1) / TB), blkE((E + TB - 1) / TB);
  dim3 blkEl((N * 32 + TB - 1) / TB);
  dim3 blkG((N + 127) / 128);
  dim3 blkNode((N + 7) / 8);

  // ---- CSR + norms (built fresh each call, shared by all 3 layers) ----
  k_zero_int<<<blkN, TB, 0, stream>>>(indeg, N);
  k_indeg<<<blkE, TB, 0, stream>>>(dst, indeg, E);
  k_dinv<<<blkN, TB, 0, stream>>>(indeg, dinv, N);
  k_scan<<<1, 1024, 0, stream>>>(indeg, off, N);
  k_copy_int<<<blkN, TB, 0, stream>>>(off, cursor, N);
  k_fill_csr<<<blkE, TB, 0, stream>>>(src, dst, cursor, csr, E);
  k_pad<<<(128 * 128 + TB - 1) / TB, TB, 0, stream>>>(Wl1, bl1, Wl2, Wl1p, bl1p, Wl2p);

  // conv1: A1 = relu(agg(x@W1) + b1)
  k_gemm128<<<blkG, TB, 0, stream>>>(x, W1, T, N);
  k_aggregate<<<blkNode, TB, 0, stream>>>(T, A1, csr, off, dinv, b1, N, 1);

  // conv2: A2 = relu(agg(A1@W2) + b2)
  k_gemm128<<<blkG, TB, 0, stream>>>(A1, W2, T, N);
  k_aggregate<<<blkNode, TB, 0, stream>>>(T, A2, csr, off, dinv, b2, N, 1);

  // conv3 (no relu): A1 = agg(A2@W3) + b3   -> h
  k_gemm128<<<blkG, TB, 0, stream>>>(A2, W3, T, N);
  k_aggregate<<<blkNode, TB, 0, stream>>>(T, A1, csr, off, dinv, b3, N, 0);

  // x1 = relu(h @ Wl1p + bl1p)
  k_gemm128<<<blkG, TB, 0, stream>>>(A1, Wl1p, A2, N);
  k_bias_act<<<blkEl, TB, 0, stream>>>(A2, bl1p, nullptr, N, 1);

  // h2 = relu(x1 @ Wl2p + bl2 + h)
  k_gemm128<<<blkG, TB, 0, stream>>>(A2, Wl2p, T, N);
  k_bias_act<<<blkEl, TB, 0, stream>>>(T, bl2, A1, N, 1);

  // head + mean pool
  k_zero<<<(3 * G + TB - 1) / TB, TB, 0, stream>>>(sums, 3 * G);
  k_head_pool<<<blkNode, TB, 0, stream>>>(T, Wl3, bl3, batch, sums, cnt, N);
  k_mean<<<(2 * G + TB - 1) / TB, TB, 0, stream>>>(sums, cnt, out, 2 * G);
}